// LinearSSM_2542620639811
// MI455X (gfx1250) — compile-verified
//
#include <hip/hip_runtime.h>

typedef __attribute__((ext_vector_type(2))) float v2f;
typedef __attribute__((ext_vector_type(8))) float v8f;
typedef __attribute__((ext_vector_type(4))) int   v4i;

#define BSZ 128
#define HH  512
#define SS  1024
#define AD  128
#define SO  (513 * 1024)   // xs stride per batch (H+1 rows of S)
#define USO (HH * AD)      // u stride per batch
#define NWG 64             // persistent grid: must be co-resident
#define NCH 36             // K-chunks of 32: ic 0..3 = u|B part, 4..35 = X|A part

// ---- gfx1250 async global->LDS (ASYNCcnt) with compile-safe fallback ------
#if defined(__AMDGCN__) && __has_builtin(__builtin_amdgcn_global_load_async_to_lds_b128)
#define HAVE_ASYNC_LDS 1
#endif

__device__ __forceinline__ void copy_b128(const float* g, float* l) {
#ifdef HAVE_ASYNC_LDS
    __builtin_amdgcn_global_load_async_to_lds_b128(
        (__attribute__((address_space(1))) v4i*)g,   // global src
        (__attribute__((address_space(3))) v4i*)l,   // LDS dst
        0, 0);
#else
    *(float4*)l = *(const float4*)g;   // sync fallback
#endif
}

// wait until at most 5 async ops outstanding (== newest chunk only; each wave
// issues exactly 5 ops per chunk, completion is in-order per the ISA)
__device__ __forceinline__ void wait_async_prev() {
#ifdef HAVE_ASYNC_LDS
#if __has_builtin(__builtin_amdgcn_s_wait_asynccnt)
    __builtin_amdgcn_s_wait_asynccnt(5);
#else
    asm volatile("s_wait_asynccnt 0x5" ::: "memory");
#endif
#endif
}

// ---------------------------------------------------------------------------
__global__ void bar_init_kernel(unsigned* bar) {
    if (threadIdx.x == 0) *bar = 0u;
}

// split device-wide barrier: arrive (release) / wait-for (acquire)
__device__ __forceinline__ void barrier_arrive(unsigned* bar, int tid) {
    __builtin_amdgcn_fence(__ATOMIC_RELEASE, "agent");
    __syncthreads();   // all waves of this wg flushed their stores
    if (tid == 0)
        __hip_atomic_fetch_add(bar, 1u, __ATOMIC_RELAXED, __HIP_MEMORY_SCOPE_AGENT);
}
__device__ __forceinline__ void barrier_waitfor(unsigned* bar, unsigned target) {
    while (__hip_atomic_load(bar, __ATOMIC_RELAXED, __HIP_MEMORY_SCOPE_AGENT) < target) {}
    __builtin_amdgcn_fence(__ATOMIC_ACQUIRE, "agent");   // global_inv vs stale WGP$
}

// ---------------------------------------------------------------------------
// Persistent fused scan: x_{t+1} = [u_t | X_t] @ [B | A]^T  (K = 128+1024)
// 64 wgs x 256 thr; wg owns 16 s-cols, wave w owns b-rows [16w,16w+16).
// Triple-buffered async LDS pipeline (distance 2); u-chunks lead each step so
// the cross-step barrier spin is overlapped with independent u@B^T work.
// ---------------------------------------------------------------------------
__global__ __launch_bounds__(256) void ssm_scan_kernel(
        const float* __restrict__ x0, const float* __restrict__ u,
        const float* __restrict__ Am, const float* __restrict__ Bm,
        float* __restrict__ xs, unsigned* __restrict__ bar) {
    __shared__ float Xs[3][128][36];   // stride 36: f4-aligned, conflict-free b64 reads
    __shared__ float As[3][16][36];

    int s0w  = blockIdx.x << 4;
    int tid  = threadIdx.x;
    int lane = tid & 31;
    int wave = tid >> 5;
    int b0   = wave << 4;
    int mrow = lane & 15;
    int khi  = (lane >> 4) << 1;    // lanes 16-31 hold K+2,K+3 (ISA f32 layout)

    // issue K-chunk ic of step t into LDS buffer p: exactly 5 async ops/wave
    auto issue_chunk = [&](int p, int ic, int t) {
        const float* xsrc;  const float* asrc;  int xstr, astr;
        if (ic < 4) {                           // u part: u[:,t,:], B
            int ka = ic << 5;
            xsrc = u + t * AD + ka;        xstr = USO;
            asrc = Bm + s0w * AD + ka;     astr = AD;
        } else {                                // X part: xs[:,t,:], A
            int kc = (ic - 4) << 5;
            xsrc = xs + t * SS + kc;       xstr = SO;
            asrc = Am + s0w * SS + kc;     astr = SS;
        }
        // M tile: 128 rows x 32 K = 1024 float4, 4 per thread (coalesced)
#pragma unroll
        for (int i = 0; i < 4; ++i) {
            int e = tid + i * 256;
            int row = e >> 3, c4 = (e & 7) << 2;
            copy_b128(xsrc + (size_t)row * xstr + c4, &Xs[p][row][c4]);
        }
        // N tile: 16 rows x 32 K = 128 float4; one op per wave (lanes 0-15)
        if (lane < 16) {
            int e = (wave << 4) + lane;
            int row = e >> 3, c4 = (e & 7) << 2;
            copy_b128(asrc + row * astr + c4, &As[p][row][c4]);
        }
    };

    // ---- prologue: xs[:,0,:] = x0, arrive, prime pipeline with u-chunks ----
    for (int e = blockIdx.x * 256 + tid; e < BSZ * SS; e += NWG * 256) {
        int b = e >> 10, s = e & 1023;
        xs[b * SO + s] = x0[e];
    }
    barrier_arrive(bar, tid);          // epoch 1: x0 visible
    issue_chunk(0, 0, 0);
    issue_chunk(1, 1, 0);

    for (int t = 0; t < HH; ++t) {
        v8f acc_a = {}, acc_b = {};    // two interleaved WMMA chains
        int p = 0, q = 2;              // ic % 3, (ic+2) % 3   (36 % 3 == 0)
        for (int ic = 0; ic < NCH; ++ic) {
            wait_async_prev();         // chunk ic landed (in-order, <=5 left)
            __syncthreads();           // ... for every wave
            if (ic == 2)               // before first X-chunk issue: sync step
                barrier_waitfor(bar, (unsigned)(NWG * (t + 1)));
            {                          // prefetch chunk ic+2 (next step / dummy)
                int icn = ic + 2, tn = t;
                if (icn >= NCH) { icn -= NCH; tn = (t + 1 < HH) ? t + 1 : t; }
                issue_chunk(q, icn, tn);
            }
#pragma unroll
            for (int ks = 0; ks < 32; ks += 8) {
                int kb0 = ks + khi, kb1 = ks + 4 + khi;
                v2f xf0 = *(const v2f*)&Xs[p][b0 + mrow][kb0];
                v2f wf0 = *(const v2f*)&As[p][mrow][kb0];
                acc_a = __builtin_amdgcn_wmma_f32_16x16x4_f32(
                    false, xf0, false, wf0, (short)0, acc_a, false, false);
                v2f xf1 = *(const v2f*)&Xs[p][b0 + mrow][kb1];
                v2f wf1 = *(const v2f*)&As[p][mrow][kb1];
                acc_b = __builtin_amdgcn_wmma_f32_16x16x4_f32(
                    false, xf1, false, wf1, (short)0, acc_b, false, false);
            }
            p = (p == 2) ? 0 : p + 1;
            q = (q == 2) ? 0 : q + 1;
        }

        // D layout: VGPR i -> M=i (lanes 0-15) / M=i+8 (lanes 16-31), N=lane&15
        v8f acc = acc_a + acc_b;
        int mbase = b0 + ((lane >> 4) << 3);
        float* op = xs + (size_t)(t + 1) * SS + s0w + mrow;
#pragma unroll
        for (int i = 0; i < 8; ++i)
            op[(size_t)(mbase + i) * SO] = acc[i];

        barrier_arrive(bar, tid);      // epoch t+2
    }
}

// ---------------------------------------------------------------------------
extern "C" void kernel_launch(void* const* d_in, const int* in_sizes, int n_in,
                              void* d_out, int out_size, void* d_ws, size_t ws_size,
                              hipStream_t stream) {
    (void)in_sizes; (void)n_in; (void)out_size; (void)ws_size;
    const float* x0 = (const float*)d_in[0];
    const float* u  = (const float*)d_in[1];
    const float* A  = (const float*)d_in[2];
    const float* B  = (const float*)d_in[3];
    float* xs = (float*)d_out;
    unsigned* bar = (unsigned*)d_ws;

    bar_init_kernel<<<1, 64, 0, stream>>>(bar);
    ssm_scan_kernel<<<NWG, 256, 0, stream>>>(x0, u, A, B, xs, bar);
}